// WarpSpatialTransformer_43722767073735
// MI455X (gfx1250) — compile-verified
//
#include <hip/hip_runtime.h>
#include <stdint.h>

#ifndef __has_builtin
#define __has_builtin(x) 0
#endif

#if __has_builtin(__builtin_amdgcn_global_load_async_to_lds_b32) && \
    __has_builtin(__builtin_amdgcn_s_wait_asynccnt)
#define WARP_USE_ASYNC 1
#else
#define WARP_USE_ASYNC 0
#endif

#if WARP_USE_ASYNC && __has_builtin(__builtin_amdgcn_global_load_async_to_lds_b128)
#define WARP_ASYNC_B128 1
#else
#define WARP_ASYNC_B128 0
#endif

namespace {
constexpr int kB = 2, kX = 160, kY = 192, kZ = 160;
constexpr int kYZ  = kY * kZ;        // 30720
constexpr int kXYZ = kX * kYZ;       // 4915200
constexpr int kRowsPerTile = 2;                       // y-rows per tile
constexpr int kTiles = 4;                             // tiles per block (pipelined)
constexpr int kRowsPerBlock = kRowsPerTile * kTiles;  // 8
constexpr int kTileVox = kRowsPerTile * kZ;           // 320 voxels / tile
constexpr int kTileTrf = kTileVox * 3;                // 960 dwords / tile (3840 B)
}  // namespace

#if WARP_USE_ASYNC
typedef __attribute__((address_space(1))) int gint;  // global ptr (AS1)
typedef __attribute__((address_space(3))) int lint;  // LDS ptr (AS3)
#endif
#if WARP_ASYNC_B128
typedef int v4i __attribute__((__vector_size__(16)));
typedef __attribute__((address_space(1))) v4i gv4i;  // global v4i* (AS1)
typedef __attribute__((address_space(3))) v4i lv4i;  // LDS v4i* (AS3)
#endif

// Per-axis clipped corners + weights, all float (corner coords are exact
// small integers). clip(floor(f)) == floor(clip(f)) for clamp-to-edge.
struct AxisF {
  float f0, f1;   // clipped floor/ceil corner coordinates (integer-valued)
  float wlo, whi; // wlo weights the floor corner (reference: w1 = loc1 - clipped)
};

__device__ __forceinline__ AxisF make_axis(float f, float maxf) {
  AxisF a;
  const float c = fminf(fmaxf(f, 0.0f), maxf);
  a.f0  = floorf(c);
  a.f1  = fminf(a.f0 + 1.0f, maxf);
  a.wlo = a.f1 - c;       // exactly 0 at the high clamp, exactly 1 at the low clamp
  a.whi = 1.0f - a.wlo;
  return a;
}

__device__ __forceinline__ void load_pair(const float* __restrict__ p, int idx,
                                          float* v) {
  __builtin_memcpy(v, p + idx, 2 * sizeof(float));  // align-4 global_load_b64
}

__global__ __launch_bounds__(kTileVox) void warp3d_kernel(
    const float* __restrict__ vol, const float* __restrict__ trf,
    float* __restrict__ out) {
  // double-buffered trf staging: 2 x 960 dwords = 7.5 KB LDS, 16-B aligned
  __shared__ __align__(16) float strf[2][kTileTrf];

  const int tx = threadIdx.x;               // z coordinate (0..159)
  const int lt = threadIdx.y * kZ + tx;     // linear id within tile (0..319)
  const int x  = blockIdx.y;                // block-uniform
  const int b  = blockIdx.z;                // block-uniform
  const int y0 = blockIdx.x * kRowsPerBlock;

  const int trfBase = ((b * kX + x) * kY + y0) * (kZ * 3);  // multiple of 480 dwords
  const int outBase = ((b * kX + x) * kY + y0) * kZ;
  const float* vb = vol + b * kXYZ;  // SGPR base for all gathers (b is uniform)

  // Stage one tile's trf block (960 contiguous dwords) into LDS buffer `parity`.
  auto issue = [&](int k, int parity) {
#if WARP_ASYNC_B128
    // 240 lanes x 16 B, all naturally aligned (trfBase*4 and tile stride are
    // multiples of 16 B). Waves 8-9 take no part; their ASYNCcnt stays 0.
    if (lt < kTileTrf / 4) {
      const uint32_t* g =
          reinterpret_cast<const uint32_t*>(trf) + trfBase + k * kTileTrf + lt * 4;
      gv4i* gp = (gv4i*)g;
      lv4i* lp = (lv4i*)&strf[parity][lt * 4];
      __builtin_amdgcn_global_load_async_to_lds_b128(gp, lp, 0, 0);
    }
#elif WARP_USE_ASYNC
    const uint32_t* g = reinterpret_cast<const uint32_t*>(trf) + trfBase + k * kTileTrf + lt;
    gint* gp = (gint*)g;
    lint* lp = (lint*)&strf[parity][lt];
    // IOFFSET is added to BOTH the global and LDS address, so one base pair
    // + 3 immediate byte offsets (320 dwords apart) covers the 960-dword tile.
    __builtin_amdgcn_global_load_async_to_lds_b32(gp, lp, 0,    0);
    __builtin_amdgcn_global_load_async_to_lds_b32(gp, lp, 1280, 0);
    __builtin_amdgcn_global_load_async_to_lds_b32(gp, lp, 2560, 0);
#else
    const int base3 = trfBase + k * kTileTrf;
    strf[parity][lt]       = trf[base3 + lt];
    strf[parity][lt + 320] = trf[base3 + lt + 320];
    strf[parity][lt + 640] = trf[base3 + lt + 640];
#endif
  };

  issue(0, 0);
  issue(1, 1);

#pragma unroll
  for (int k = 0; k < kTiles; ++k) {
    const int p = k & 1;
#if WARP_ASYNC_B128
    if (k + 1 < kTiles) {
      __builtin_amdgcn_s_wait_asynccnt(1);  // this wave's tile-k copy done
    } else {
      __builtin_amdgcn_s_wait_asynccnt(0);  // last tile: drain
    }
#elif WARP_USE_ASYNC
    if (k + 1 < kTiles) {
      __builtin_amdgcn_s_wait_asynccnt(3);  // this wave's tile-k copies done
    } else {
      __builtin_amdgcn_s_wait_asynccnt(0);  // last tile: drain
    }
#endif
    __syncthreads();  // all waves' tile-k copies visible

    const float dx = strf[p][3 * lt + 0];
    const float dy = strf[p][3 * lt + 1];
    const float dz = strf[p][3 * lt + 2];

    const int y = y0 + k * kRowsPerTile + (int)threadIdx.y;

    const AxisF ax = make_axis((float)x  + dx, (float)(kX - 1));
    const AxisF ay = make_axis((float)y  + dy, (float)(kY - 1));
    const AxisF az = make_axis((float)tx + dz, (float)(kZ - 1));

    // z-pair base zb = min(z0, Z-2): [zb, zb+1] always in-bounds. No select is
    // needed: if z0 was clamped to Z-1 then az.wlo == 0 exactly, so the
    // pair[0] (= v[Z-2]) term vanishes and pair[1] (= v[Z-1]) carries whi == 1.
    const float zbf = fminf(az.f0, (float)(kZ - 2));

    // Exact float fma addressing: every intermediate is an integer < 2^24.
    const float my0 = fmaf(ay.f0, (float)kZ, zbf);
    const float my1 = fmaf(ay.f1, (float)kZ, zbf);
    const int i00 = (int)fmaf(ax.f0, (float)kYZ, my0);
    const int i01 = (int)fmaf(ax.f0, (float)kYZ, my1);
    const int i10 = (int)fmaf(ax.f1, (float)kYZ, my0);
    const int i11 = (int)fmaf(ax.f1, (float)kYZ, my1);

    float p00[2], p01[2], p10[2], p11[2];
    load_pair(vb, i00, p00);
    load_pair(vb, i01, p01);
    load_pair(vb, i10, p10);
    load_pair(vb, i11, p11);

    const float c00 = fmaf(p00[0], az.wlo, p00[1] * az.whi);
    const float c01 = fmaf(p01[0], az.wlo, p01[1] * az.whi);
    const float c10 = fmaf(p10[0], az.wlo, p10[1] * az.whi);
    const float c11 = fmaf(p11[0], az.wlo, p11[1] * az.whi);
    const float c0  = fmaf(c00, ay.wlo, c01 * ay.whi);
    const float c1  = fmaf(c10, ay.wlo, c11 * ay.whi);
    const float res = fmaf(c0,  ax.wlo, c1  * ax.whi);

    // out is write-once / never re-read: non-temporal keeps vol resident in L2
    __builtin_nontemporal_store(res, &out[outBase + k * kTileVox + lt]);

    __syncthreads();  // all waves finished reading buffer p
    if (k + 2 < kTiles) issue(k + 2, p);  // safe to overwrite buffer p
  }
}

extern "C" void kernel_launch(void* const* d_in, const int* in_sizes, int n_in,
                              void* d_out, int out_size, void* d_ws, size_t ws_size,
                              hipStream_t stream) {
  const float* vol = (const float*)d_in[0];  // [2,160,192,160,1] fp32
  const float* trf = (const float*)d_in[1];  // [2,160,192,160,3] fp32
  float* out = (float*)d_out;                // [2,160,192,160,1] fp32
  (void)in_sizes; (void)n_in; (void)out_size; (void)d_ws; (void)ws_size;

  dim3 grid(kY / kRowsPerBlock, kX, kB);   // (24, 160, 2)
  dim3 block(kZ, kRowsPerTile);            // (160, 2) = 320 threads, 10 waves
  warp3d_kernel<<<grid, block, 0, stream>>>(vol, trf, out);
}